// JKNetMaxpool_19344532701769
// MI455X (gfx1250) — compile-verified
//
#include <hip/hip_runtime.h>

#define FEAT 128
#define NCLS 40
#define NLAYERS 4

typedef __attribute__((ext_vector_type(2))) float v2f;
typedef __attribute__((ext_vector_type(8))) float v8f;

#define WMMA_F32X4(av, bv, cv) \
    __builtin_amdgcn_wmma_f32_16x16x4_f32(false, (av), false, (bv), (short)0, (cv), false, false)

// ---------------- degree / normalization ----------------

__global__ void init_deg_kernel(float* __restrict__ deg, int n) {
    int i = blockIdx.x * blockDim.x + threadIdx.x;
    if (i < n) deg[i] = 1.0f;   // self-loop contribution
}

__global__ void accum_deg_kernel(const int* __restrict__ dst, float* __restrict__ deg, int e) {
    int i = blockIdx.x * blockDim.x + threadIdx.x;
    if (i < e) atomicAdd(&deg[dst[i]], 1.0f);
}

__global__ void dinv_kernel(float* __restrict__ deg_dinv, int n) {
    int i = blockIdx.x * blockDim.x + threadIdx.x;
    if (i < n) {
        float d = deg_dinv[i];
        deg_dinv[i] = (d > 0.0f) ? rsqrtf(d) : 0.0f;
    }
}

// ---------------- specialized WMMA GEMM: D[nrows,128] = A[nrows,128] @ W[128,128] ------------
// W staged in LDS K-pair-interleaved so each B operand pair {W[k][c],W[k+1][c]} is one
// ds_load_b64 into an even-aligned VGPR pair (no repack moves). Each wave computes a 16x128
// strip: 8 accumulators, 8 back-to-back V_WMMA_F32_16X16X4_F32 per K-step off one v2f A load.
// A-matrix VGPR layout (16x4 f32): lanes 0-15 hold K=k,k+1; lanes 16-31 hold K=k+2,k+3; M=lane&15.
// B-matrix (4x16 f32): mirrored K split across lane halves; N=lane&15.
// C/D (16x16 f32, 8 VGPRs): VGPR v = row v (lanes 0-15) / row v+8 (lanes 16-31).

__global__ __launch_bounds__(256)
void gemm128_wmma_kernel(const float* __restrict__ A, const float* __restrict__ W,
                         float* __restrict__ D, int nrows) {
    // layout: w_s[(kk*FEAT + c)*2 + 0] = W[2kk][c]; ...*2 + 1] = W[2kk+1][c]   (kk = k/2)
    __shared__ float w_s[FEAT * FEAT];   // 64 KB

    // cooperative interleaving load: 2048 (kk, c4) units, 8 per thread
    for (int i = threadIdx.x; i < (FEAT / 2) * (FEAT / 4); i += 256) {
        int kk = i >> 5;              // 0..63
        int c4 = (i & 31) << 2;       // column, step 4
        float4 r0 = *(const float4*)(W + (size_t)(2 * kk)     * FEAT + c4);
        float4 r1 = *(const float4*)(W + (size_t)(2 * kk + 1) * FEAT + c4);
        float* o = w_s + ((size_t)kk * FEAT + c4) * 2;
        *(float4*)(o + 0) = make_float4(r0.x, r1.x, r0.y, r1.y);
        *(float4*)(o + 4) = make_float4(r0.z, r1.z, r0.w, r1.w);
    }
    __syncthreads();

    int wave = threadIdx.x >> 5;
    int lane = threadIdx.x & 31;
    int rowTiles = (nrows + 15) >> 4;
    int rowTile = blockIdx.x * 8 + wave;
    if (rowTile >= rowTiles) return;     // wave-uniform

    int half = lane >> 4;                // 0: lanes 0-15, 1: lanes 16-31
    int r    = lane & 15;
    int row0 = rowTile << 4;
    int aRow = min(row0 + r, nrows - 1); // clamp, no divergence
    const float* aPtr  = A + (size_t)aRow * FEAT + 2 * half;
    const float* bBase = w_s + ((size_t)half * FEAT + r) * 2;

    v8f acc[8];
    #pragma unroll
    for (int t = 0; t < 8; ++t)
        acc[t] = (v8f){0.f, 0.f, 0.f, 0.f, 0.f, 0.f, 0.f, 0.f};

    #pragma unroll 2
    for (int k = 0; k < FEAT; k += 4) {
        v2f av = *(const v2f*)(aPtr + k);
        const v2f* bp = (const v2f*)(bBase + (size_t)k * FEAT);  // (k/2 + half) pair row
        #pragma unroll
        for (int t = 0; t < 8; ++t) {
            v2f bv = bp[t * 16];                                 // ds_load_b64, imm offset t*128B
            acc[t] = WMMA_F32X4(av, bv, acc[t]);
        }
    }

    if (row0 + 16 <= nrows) {
        // full tile: unconditional, clause-able stores
        float* dBase = D + ((size_t)row0 + 8 * half) * FEAT + r;
        #pragma unroll
        for (int v = 0; v < 8; ++v) {
            #pragma unroll
            for (int t = 0; t < 8; ++t)
                dBase[(size_t)v * FEAT + t * 16] = acc[t][v];
        }
    } else {
        #pragma unroll
        for (int v = 0; v < 8; ++v) {
            int rr = row0 + v + 8 * half;
            if (rr < nrows) {
                #pragma unroll
                for (int t = 0; t < 8; ++t)
                    D[(size_t)rr * FEAT + t * 16 + r] = acc[t][v];
            }
        }
    }
}

// ---------------- generic small-N WMMA GEMM (classifier, ncols=40) ----------------

__global__ __launch_bounds__(256)
void gemm_small_kernel(const float* __restrict__ A, const float* __restrict__ B,
                       const float* __restrict__ bias, float* __restrict__ D,
                       int nrows, int ncols) {
    int wave = threadIdx.x >> 5;
    int lane = threadIdx.x & 31;
    int colTiles = (ncols + 15) >> 4;
    int rowTiles = (nrows + 15) >> 4;
    long tile = (long)blockIdx.x * 8 + wave;
    if (tile >= (long)rowTiles * colTiles) return;   // wave-uniform
    int rowTile = (int)(tile / colTiles);
    int colTile = (int)(tile % colTiles);
    int row0 = rowTile << 4;
    int col0 = colTile << 4;

    int half = lane >> 4;
    int r    = lane & 15;
    int aRow      = min(row0 + r, nrows - 1);        // clamp, unconditional loads
    int bColStore = col0 + r;
    int bCol      = min(bColStore, ncols - 1);

    const float* aPtr = A + (size_t)aRow * FEAT + 2 * half;
    const float* bPtr = B + (size_t)(2 * half) * ncols + bCol;

    v8f acc = (v8f){0.f, 0.f, 0.f, 0.f, 0.f, 0.f, 0.f, 0.f};
    #pragma unroll 4
    for (int k = 0; k < FEAT; k += 4) {
        v2f av = *(const v2f*)(aPtr + k);
        v2f bv;
        bv.x = bPtr[(size_t)k * ncols];
        bv.y = bPtr[(size_t)(k + 1) * ncols];
        acc = WMMA_F32X4(av, bv, acc);
    }

    if (bColStore < ncols) {
        float badd = bias ? bias[bColStore] : 0.0f;
        #pragma unroll
        for (int v = 0; v < 8; ++v) {
            int rr = row0 + v + 8 * half;
            if (rr < nrows)
                D[(size_t)rr * ncols + bColStore] = acc[v] + badd;
        }
    }
}

// ---------------- self-loop init: agg = t * dinv^2 ----------------

__global__ void self_loop_kernel(const float* __restrict__ T, const float* __restrict__ dinv,
                                 float* __restrict__ agg, int n) {
    int g = blockIdx.x * blockDim.x + threadIdx.x;   // n*32 threads, float4 each
    int node = g >> 5;
    int q = g & 31;
    if (node >= n) return;
    float s = dinv[node];
    s = s * s;
    float4 v = *((const float4*)(T + (size_t)node * FEAT) + q);
    float4 o = make_float4(v.x * s, v.y * s, v.z * s, v.w * s);
    *((float4*)(agg + (size_t)node * FEAT) + q) = o;
}

// ---------------- edge scatter: agg[dst] += t[src] * dinv[src]*dinv[dst] ----------------
// One wave per edge: 32 lanes x float4 = coalesced 512B row gather + 4 f32 atomics/lane.

__global__ void scatter_kernel(const int* __restrict__ src, const int* __restrict__ dst,
                               const float* __restrict__ dinv, const float* __restrict__ T,
                               float* __restrict__ agg, int ne) {
    int g = blockIdx.x * blockDim.x + threadIdx.x;
    int e = g >> 5;
    int q = g & 31;
    if (e >= ne) return;
    int s = src[e];
    int d = dst[e];
    float norm = dinv[s] * dinv[d];
    float4 v = *((const float4*)(T + (size_t)s * FEAT) + q);
    float* p = agg + (size_t)d * FEAT + q * 4;
    atomicAdd(p + 0, v.x * norm);
    atomicAdd(p + 1, v.y * norm);
    atomicAdd(p + 2, v.z * norm);
    atomicAdd(p + 3, v.w * norm);
}

// ---------------- bias + relu + JK running max; h written in place into agg ----------------

__global__ void finish_kernel(float* __restrict__ agg, const float* __restrict__ bias,
                              float* __restrict__ hmax, int n, int first) {
    int g = blockIdx.x * blockDim.x + threadIdx.x;
    int node = g >> 5;
    int q = g & 31;
    if (node >= n) return;
    float4* ap = (float4*)(agg + (size_t)node * FEAT) + q;
    float4 v = *ap;
    float4 b = *((const float4*)bias + q);
    float4 h = make_float4(fmaxf(v.x + b.x, 0.f), fmaxf(v.y + b.y, 0.f),
                           fmaxf(v.z + b.z, 0.f), fmaxf(v.w + b.w, 0.f));
    *ap = h;
    float4* mp = (float4*)(hmax + (size_t)node * FEAT) + q;
    if (first) {
        *mp = h;
    } else {
        float4 m = *mp;
        m.x = fmaxf(m.x, h.x);
        m.y = fmaxf(m.y, h.y);
        m.z = fmaxf(m.z, h.z);
        m.w = fmaxf(m.w, h.w);
        *mp = m;
    }
}

// ---------------- host ----------------

static inline int cdivi(long a, long b) { return (int)((a + b - 1) / b); }

extern "C" void kernel_launch(void* const* d_in, const int* in_sizes, int n_in,
                              void* d_out, int out_size, void* d_ws, size_t ws_size,
                              hipStream_t stream) {
    const float* x  = (const float*)d_in[0];
    const int*   ei = (const int*)d_in[1];
    const float* W0 = (const float*)d_in[2];
    const float* b0 = (const float*)d_in[3];
    const float* Wh = (const float*)d_in[4];
    const float* bh = (const float*)d_in[5];
    const float* Wl = (const float*)d_in[6];
    const float* bl = (const float*)d_in[7];

    const int n = in_sizes[0] / FEAT;      // 100000 nodes
    const int e = in_sizes[1] / 2;         // 1.6M edges
    const int* srcp = ei;
    const int* dstp = ei + e;

    float* ws   = (float*)d_ws;
    float* dinv = ws;                                   // n floats (padded)
    float* bufA = ws + (((long)n + 255) & ~255L);       // n*128: agg / current h
    float* bufB = bufA + (size_t)n * FEAT;              // n*128: t = h@W
    float* hmax = bufB + (size_t)n * FEAT;              // n*128: JK max
    float* outp = (float*)d_out;

    // --- normalization: dinv = rsqrt(in_degree + 1) ---
    init_deg_kernel<<<cdivi(n, 256), 256, 0, stream>>>(dinv, n);
    accum_deg_kernel<<<cdivi(e, 256), 256, 0, stream>>>(dstp, dinv, e);
    dinv_kernel<<<cdivi(n, 256), 256, 0, stream>>>(dinv, n);

    const int rowTiles = cdivi(n, 16);

    const float* hcur = x;
    for (int layer = 0; layer < NLAYERS; ++layer) {
        const float* W = (layer == 0) ? W0 : Wh + (size_t)(layer - 1) * FEAT * FEAT;
        const float* b = (layer == 0) ? b0 : bh + (size_t)(layer - 1) * FEAT;

        // t = hcur @ W  (W cached in LDS, 8 WMMA-deep per wave)
        gemm128_wmma_kernel<<<cdivi(rowTiles, 8), 256, 0, stream>>>(hcur, W, bufB, n);

        self_loop_kernel<<<cdivi((long)n * 32, 256), 256, 0, stream>>>(bufB, dinv, bufA, n);
        scatter_kernel<<<cdivi((long)e * 32, 256), 256, 0, stream>>>(srcp, dstp, dinv, bufB, bufA, e);
        finish_kernel<<<cdivi((long)n * 32, 256), 256, 0, stream>>>(bufA, b, hmax, n, layer == 0);
        hcur = bufA;
    }

    // --- classifier: out = hmax @ Wl + bl ---
    long ftiles = (long)rowTiles * cdivi(NCLS, 16);
    gemm_small_kernel<<<cdivi(ftiles, 8), 256, 0, stream>>>(hmax, Wl, bl, outp, n, NCLS);
}